// HypergraphScatteringNet_51994874085933
// MI455X (gfx1250) — compile-verified
//
#include <hip/hip_runtime.h>

// ---------------------------------------------------------------------------
// HypergraphScatteringNet for MI455X (gfx1250, wave32)
//   phase 1: degree counts + weighted node degree (atomics, one pass)
//   phase 2: exclusive prefix scans (LDS) -> CSR offsets (by edge, by node)
//   phase 3: CSR fill with fused per-incidence coefficient c = Dv_inv*Dhe_inv
//   phase 4: 16 diffusion rounds, gather-only (L2-resident), wave-per-row
//   phase 5: wavelet einsum via V_WMMA_F32_16X16X4_F32 (K=17 padded to 20)
// ---------------------------------------------------------------------------

typedef __attribute__((ext_vector_type(2))) float v2f;
typedef __attribute__((ext_vector_type(8))) float v8f;

// ---------------------------- utility kernels ------------------------------

__global__ void zero_i32(int* __restrict__ p, long n) {
  long i = (long)blockIdx.x * blockDim.x + threadIdx.x;
  if (i < n) p[i] = 0;
}

__global__ void count_deg(const int* __restrict__ node_idx,
                          const int* __restrict__ edge_idx,
                          const float* __restrict__ w,
                          int* __restrict__ cnt_n, int* __restrict__ cnt_e,
                          float* __restrict__ dv, int nnz) {
  int i = blockIdx.x * blockDim.x + threadIdx.x;
  if (i >= nnz) return;
  int n = node_idx[i], e = edge_idx[i];
  atomicAdd(&cnt_n[n], 1);
  atomicAdd(&cnt_e[e], 1);
  atomicAdd(&dv[n], w[e]);
}

// exclusive scan, 1024 elements per block (256 threads x 4 items), LDS scan
__global__ void scan_block(const int* __restrict__ in, int* __restrict__ out,
                           int* __restrict__ bsums, int n) {
  __shared__ int sh[256];
  int t = threadIdx.x;
  int base = blockIdx.x * 1024 + t * 4;
  int v0 = (base + 0 < n) ? in[base + 0] : 0;
  int v1 = (base + 1 < n) ? in[base + 1] : 0;
  int v2 = (base + 2 < n) ? in[base + 2] : 0;
  int v3 = (base + 3 < n) ? in[base + 3] : 0;
  int s = v0 + v1 + v2 + v3;
  sh[t] = s;
  __syncthreads();
  for (int off = 1; off < 256; off <<= 1) {
    int x = (t >= off) ? sh[t - off] : 0;
    __syncthreads();
    sh[t] += x;
    __syncthreads();
  }
  int run = sh[t] - s;  // exclusive prefix for this thread's 4 items
  if (base + 0 < n) out[base + 0] = run;  run += v0;
  if (base + 1 < n) out[base + 1] = run;  run += v1;
  if (base + 2 < n) out[base + 2] = run;  run += v2;
  if (base + 3 < n) out[base + 3] = run;
  if (t == 255 && bsums) bsums[blockIdx.x] = sh[255];
}

__global__ void scan_add(int* __restrict__ out, const int* __restrict__ boff, int n) {
  int i = blockIdx.x * blockDim.x + threadIdx.x;
  if (i < n) out[i] += boff[i >> 10];
}

__global__ void set_tail(int* __restrict__ p, int v) { *p = v; }

__global__ void invert_deg(const float* __restrict__ dv, const int* __restrict__ cnt_e,
                           float* __restrict__ dvi, float* __restrict__ dhei,
                           int n, int e) {
  int i = blockIdx.x * blockDim.x + threadIdx.x;
  if (i < n) { float d = dv[i];    dvi[i]  = (d > 0.f) ? 1.f / d : 0.f; }
  if (i < e) { int   c = cnt_e[i]; dhei[i] = (c > 0)   ? 1.f / (float)c : 0.f; }
}

__global__ void fill_csr(const int* __restrict__ node_idx, const int* __restrict__ edge_idx,
                         const float* __restrict__ dvi, const float* __restrict__ dhei,
                         const int* __restrict__ n_offs, const int* __restrict__ e_offs,
                         int* __restrict__ cur_n, int* __restrict__ cur_e,
                         int* __restrict__ n_edges, float* __restrict__ n_coef,
                         int* __restrict__ e_nodes, float* __restrict__ e_coef,
                         int nnz) {
  int i = blockIdx.x * blockDim.x + threadIdx.x;
  if (i >= nnz) return;
  int n = node_idx[i], e = edge_idx[i];
  float cf = dvi[n] * dhei[e];           // same factor both directions
  int pe = e_offs[e] + atomicAdd(&cur_e[e], 1);
  e_nodes[pe] = n;  e_coef[pe] = cf;
  int pn = n_offs[n] + atomicAdd(&cur_n[n], 1);
  n_edges[pn] = e;  n_coef[pn] = cf;
}

// ------------------------ diffusion gather pass ----------------------------
// One wave per destination row; lane owns 4 features (float4). Entirely
// gather-based: src rows (<=25.6MB) stay resident in L2. Uniform CSR loads
// are scalarized via readfirstlane on the wave id.
__global__ void gather_pass(const float* __restrict__ src,
                            const int* __restrict__ offs,
                            const int* __restrict__ idx,
                            const float* __restrict__ coef,
                            float* __restrict__ dst, int ndst) {
  int wave = (int)((blockIdx.x * blockDim.x + threadIdx.x) >> 5);
  wave = __builtin_amdgcn_readfirstlane(wave);   // provably wave-uniform
  int lane = threadIdx.x & 31;
  if (wave >= ndst) return;
  int beg = offs[wave];
  int end = offs[wave + 1];
  const int fo = lane * 4;
  float4 acc = make_float4(0.f, 0.f, 0.f, 0.f);
  for (int p = beg; p < end; ++p) {
    int s = idx[p];
    float cf = coef[p];
    if (p + 1 < end) {
      // hide L2 gather latency for the next row (global_prefetch_b8)
      __builtin_prefetch(src + (size_t)idx[p + 1] * 128 + fo, 0, 1);
    }
    const float4 v = *reinterpret_cast<const float4*>(src + (size_t)s * 128 + fo);
    acc.x += cf * v.x; acc.y += cf * v.y; acc.z += cf * v.z; acc.w += cf * v.w;
  }
  *reinterpret_cast<float4*>(dst + (size_t)wave * 128 + fo) = acc;
}

// ------------------------- WMMA wavelet combine ----------------------------
// out[ent, s*128 + f] = sum_{l<17} W[s,l] * stack[l][ent][f]
// A = W (16x4 tile, rows 6..15 zero), B = stack slab (4x16), C/D = 16x16 f32.
// K = 17 padded to 20 -> 5 chained V_WMMA_F32_16X16X4_F32 per feature tile.
__global__ void wmma_combine(const float* __restrict__ stack,  // [17][count][128]
                             const float* __restrict__ wav,    // [6][17]
                             float* __restrict__ out,          // [count][768]
                             int count) {
  int wave = (int)((blockIdx.x * blockDim.x + threadIdx.x) >> 5);
  wave = __builtin_amdgcn_readfirstlane(wave);
  int lane = threadIdx.x & 31;
  int ntiles = count * 8;                 // 8 feature tiles of 16 per entity
  if (wave >= ntiles) return;             // wave-uniform: EXEC stays all-ones
  int ent = wave >> 3;
  int f0  = (wave & 7) << 4;
  int m     = lane & 15;                  // A row (s) / B col (f) index
  int khalf = lane >> 4;                  // K sub-pair select (ISA 16x4 layout)

  // A fragments (wavelet), zero-padded in M (s>=6) and K (l>=17)
  v2f a[5];
#pragma unroll
  for (int k = 0; k < 5; ++k) {
    int l0 = 4 * k + 2 * khalf;
    int mc = m < 6 ? m : 5;
    int lc0 = l0 < 17 ? l0 : 16;
    int lc1 = (l0 + 1) < 17 ? (l0 + 1) : 16;
    float mk0 = (m < 6 && l0     < 17) ? 1.f : 0.f;
    float mk1 = (m < 6 && (l0+1) < 17) ? 1.f : 0.f;
    a[k][0] = wav[mc * 17 + lc0] * mk0;
    a[k][1] = wav[mc * 17 + lc1] * mk1;
  }

  const size_t lvl = (size_t)count * 128;            // level stride
  const float* base = stack + (size_t)ent * 128 + f0 + m;
  v8f c = {};
#pragma unroll
  for (int k = 0; k < 5; ++k) {
    int l0 = 4 * k + 2 * khalf;
    int lc0 = l0 < 17 ? l0 : 16;
    int lc1 = (l0 + 1) < 17 ? (l0 + 1) : 16;
    float b0 = base[(size_t)lc0 * lvl] * ((l0     < 17) ? 1.f : 0.f);
    float b1 = base[(size_t)lc1 * lvl] * (((l0+1) < 17) ? 1.f : 0.f);
    v2f b; b[0] = b0; b[1] = b1;
    c = __builtin_amdgcn_wmma_f32_16x16x4_f32(false, a[k], false, b,
                                              (short)0, c, false, false);
  }

  // D layout: lanes 0-15 hold cols N=lane, VGPR j -> row M=j (s index).
  // Only s=0..5 are live; lanes 16-31 carry rows 8..15 (all zero) -> skip.
  if (khalf == 0) {
    float* o = out + (size_t)ent * 768 + f0 + m;
#pragma unroll
    for (int j = 0; j < 6; ++j) o[j * 128] = c[j];
  }
}

// ------------------------------- host side ---------------------------------

static inline char* bump(char*& p, size_t bytes) {
  char* r = p;
  p += (bytes + 255) & ~(size_t)255;
  return r;
}

extern "C" void kernel_launch(void* const* d_in, const int* in_sizes, int n_in,
                              void* d_out, int out_size, void* d_ws, size_t ws_size,
                              hipStream_t stream) {
  const float* x    = (const float*)d_in[0];
  const int*   hei  = (const int*)d_in[1];
  const float* w    = (const float*)d_in[2];
  const float* attr = (const float*)d_in[3];
  const float* wav  = (const float*)d_in[4];

  const int F   = 128;
  const int N   = in_sizes[0] / F;     // 50000
  const int NNZ = in_sizes[1] / 2;     // 1000000
  const int E   = in_sizes[2];         // 5000
  const int LV  = 17;                  // levels + 1
  const int SF  = 6 * F;               // 768 output features

  const int* node_idx = hei;
  const int* edge_idx = hei + NNZ;

  // -------- workspace layout --------
  char* p = (char*)d_ws;
  float* node_stack = (float*)bump(p, (size_t)LV * N * F * 4);
  float* edge_stack = (float*)bump(p, (size_t)LV * E * F * 4);
  int*   e_nodes    = (int*)  bump(p, (size_t)NNZ * 4);
  float* e_coef     = (float*)bump(p, (size_t)NNZ * 4);
  int*   n_edges    = (int*)  bump(p, (size_t)NNZ * 4);
  float* n_coef     = (float*)bump(p, (size_t)NNZ * 4);
  int*   n_offs     = (int*)  bump(p, (size_t)(N + 1) * 4);
  int*   e_offs     = (int*)  bump(p, (size_t)(E + 1) * 4);
  int*   zbeg       = (int*)p;       // everything below gets zeroed in one shot
  int*   cnt_n      = (int*)  bump(p, (size_t)N * 4);
  int*   cnt_e      = (int*)  bump(p, (size_t)E * 4);
  int*   cur_n      = (int*)  bump(p, (size_t)N * 4);
  int*   cur_e      = (int*)  bump(p, (size_t)E * 4);
  float* dv         = (float*)bump(p, (size_t)N * 4);
  int*   zend       = (int*)p;
  float* dvi        = (float*)bump(p, (size_t)N * 4);
  float* dhei       = (float*)bump(p, (size_t)E * 4);
  int*   bsums      = (int*)  bump(p, 2048 * 4);
  int*   bsums2     = (int*)  bump(p, 2048 * 4);

  const int B = 256;

  // phase 0: zero counters/cursors/weighted degree
  long zwords = zend - zbeg;
  zero_i32<<<(int)((zwords + B - 1) / B), B, 0, stream>>>(zbeg, zwords);

  // phase 1: degrees
  count_deg<<<(NNZ + B - 1) / B, B, 0, stream>>>(node_idx, edge_idx, w,
                                                 cnt_n, cnt_e, dv, NNZ);

  // phase 2: exclusive scans -> CSR offsets
  int nb_n = (N + 1023) / 1024;
  scan_block<<<nb_n, 256, 0, stream>>>(cnt_n, n_offs, bsums, N);
  scan_block<<<1, 256, 0, stream>>>(bsums, bsums2, (int*)nullptr, nb_n);
  scan_add<<<(N + B - 1) / B, B, 0, stream>>>(n_offs, bsums2, N);
  set_tail<<<1, 1, 0, stream>>>(n_offs + N, NNZ);

  int nb_e = (E + 1023) / 1024;
  scan_block<<<nb_e, 256, 0, stream>>>(cnt_e, e_offs, bsums, E);
  scan_block<<<1, 256, 0, stream>>>(bsums, bsums2, (int*)nullptr, nb_e);
  scan_add<<<(E + B - 1) / B, B, 0, stream>>>(e_offs, bsums2, E);
  set_tail<<<1, 1, 0, stream>>>(e_offs + E, NNZ);

  // phase 3: normalizers + CSR fill with fused coefficient
  int NE = N > E ? N : E;
  invert_deg<<<(NE + B - 1) / B, B, 0, stream>>>(dv, cnt_e, dvi, dhei, N, E);
  fill_csr<<<(NNZ + B - 1) / B, B, 0, stream>>>(node_idx, edge_idx, dvi, dhei,
                                                n_offs, e_offs, cur_n, cur_e,
                                                n_edges, n_coef, e_nodes, e_coef, NNZ);

  // level 0 snapshots
  hipMemcpyAsync(node_stack, x,    (size_t)N * F * 4, hipMemcpyDeviceToDevice, stream);
  hipMemcpyAsync(edge_stack, attr, (size_t)E * F * 4, hipMemcpyDeviceToDevice, stream);

  // phase 4: 16 diffusion rounds (all gathers; L2-resident working set)
  for (int l = 1; l < LV; ++l) {
    const float* nsrc = node_stack + (size_t)(l - 1) * N * F;
    float*       edst = edge_stack + (size_t)l * E * F;
    gather_pass<<<(E * 32 + B - 1) / B, B, 0, stream>>>(nsrc, e_offs, e_nodes,
                                                        e_coef, edst, E);
    float*       ndst = node_stack + (size_t)l * N * F;
    gather_pass<<<(N * 32 + B - 1) / B, B, 0, stream>>>(edst, n_offs, n_edges,
                                                        n_coef, ndst, N);
  }

  // phase 5: wavelet einsum via f32 WMMA
  float* out_nodes = (float*)d_out;
  float* out_edges = out_nodes + (size_t)N * SF;
  wmma_combine<<<(N * 8 * 32 + B - 1) / B, B, 0, stream>>>(node_stack, wav,
                                                           out_nodes, N);
  wmma_combine<<<(E * 8 * 32 + B - 1) / B, B, 0, stream>>>(edge_stack, wav,
                                                           out_edges, E);
}